// FeatureAndClassWeightedBCELoss_76690936037931
// MI455X (gfx1250) — compile-verified
//
#include <hip/hip_runtime.h>
#include <hip/hip_bf16.h>

// ---------------------------------------------------------------------------
// FeatureAndClassWeightedBCELoss — memory-bound streaming reduction.
// 268 MB in -> 1 float out. Strategy: b128 non-temporal coalesced loads,
// grid-stride accumulation, WMMA-f32-based wave reduction (v_wmma_f32_16x16x4_f32),
// deterministic two-pass final reduce.
// ---------------------------------------------------------------------------

typedef __attribute__((ext_vector_type(4))) float f32x4;
typedef __attribute__((ext_vector_type(4))) int   i32x4;
typedef __attribute__((ext_vector_type(2))) float v2f;
typedef __attribute__((ext_vector_type(8))) float v8f;

#define NBLOCKS 4096
#define NTHREADS 256
#define NWAVES (NTHREADS / 32)

// Gaussian weight constant: (1 / (20*sqrt(2*pi))) * 100
#define PDF_SCALE 1.9947114f

// Per-sample weighted CE loss.
__device__ __forceinline__ float sample_loss(float x0, float x1, int t, float fw) {
    // 2-class CE: -logp[t] = softplus(x_other - x_t), computed stably.
    float xt = t ? x1 : x0;
    float xo = t ? x0 : x1;
    float d  = xo - xt;
    float loss = fmaxf(d, 0.0f) + __logf(1.0f + __expf(-fabsf(d)));
    float cw = t ? 2.0f : 1.0f;
    // Gaussian feature weight
    float z   = (fw - 70.0f) * 0.05f;          // /sigma (=20)
    float pdf = __expf(-0.5f * z * z) * PDF_SCALE;
    float w   = (fw < 70.0f) ? pdf : 1.0f;
    w = fmaxf(w, 1e-6f);
    return loss * (cw * w);
}

// Wave32 all-lane sum using V_WMMA_F32_16X16X4_F32.
// A (16x4 f32, 2 VGPRs): VGPR0 = per-lane partial (lane m -> A[m,0], lane m+16
// -> A[m,2]), VGPR1 = 0. B = ones 4x16, C = 0.
// => D[m,n] = partial[m] + partial[m+16].
// Lane n (n<16) holds D[0..7,n] in d[0..7]; lane n+16 holds D[8..15,n].
// Sum the 8 D regs per lane, then fold halves with shfl_xor(16).
__device__ __forceinline__ float wave_reduce_f32(float v) {
    v2f a;
    a.x = v;
    a.y = 0.0f;
    v2f b;
    b.x = 1.0f;
    b.y = 1.0f;
    v8f c = {};
    v8f d = __builtin_amdgcn_wmma_f32_16x16x4_f32(
        /*neg_a=*/false, a, /*neg_b=*/false, b,
        /*c_mod=*/(short)0, c, /*reuse_a=*/false, /*reuse_b=*/false);
    float s = ((d[0] + d[1]) + (d[2] + d[3])) + ((d[4] + d[5]) + (d[6] + d[7]));
    s += __shfl_xor(s, 16, 32);
    return s;
}

__global__ void __launch_bounds__(NTHREADS)
loss_partial_kernel(const float* __restrict__ logits,
                    const int* __restrict__ targets,
                    const float* __restrict__ fw,
                    float* __restrict__ block_partials,
                    int n_groups) {  // n_groups = B/4
    const f32x4* __restrict__ lg4 = (const f32x4*)logits;   // 2 per group
    const i32x4* __restrict__ tg4 = (const i32x4*)targets;  // 1 per group
    const f32x4* __restrict__ fw4 = (const f32x4*)fw;       // 1 per group

    float acc = 0.0f;
    int stride = gridDim.x * blockDim.x;
    for (int g = blockIdx.x * blockDim.x + threadIdx.x; g < n_groups; g += stride) {
        // Streamed once; >L2 footprint -> non-temporal b128 loads.
        f32x4 l0 = __builtin_nontemporal_load(lg4 + 2 * (long long)g);
        f32x4 l1 = __builtin_nontemporal_load(lg4 + 2 * (long long)g + 1);
        i32x4 t  = __builtin_nontemporal_load(tg4 + g);
        f32x4 w  = __builtin_nontemporal_load(fw4 + g);
        acc += sample_loss(l0.x, l0.y, t.x, w.x);
        acc += sample_loss(l0.z, l0.w, t.y, w.y);
        acc += sample_loss(l1.x, l1.y, t.z, w.z);
        acc += sample_loss(l1.z, l1.w, t.w, w.w);
    }

    // Wave-level reduce via WMMA, then block-level via LDS.
    float wsum = wave_reduce_f32(acc);

    __shared__ float lds_sums[NWAVES];
    int lane = threadIdx.x & 31;
    int wid  = threadIdx.x >> 5;
    if (lane == 0) lds_sums[wid] = wsum;
    __syncthreads();
    if (threadIdx.x == 0) {
        float s = 0.0f;
        #pragma unroll
        for (int i = 0; i < NWAVES; ++i) s += lds_sums[i];
        block_partials[blockIdx.x] = s;
    }
}

__global__ void __launch_bounds__(NTHREADS)
loss_finalize_kernel(const float* __restrict__ block_partials,
                     int n_partials,
                     float* __restrict__ out,
                     float inv_b) {
    float acc = 0.0f;
    for (int i = threadIdx.x; i < n_partials; i += NTHREADS)
        acc += block_partials[i];

    float wsum = wave_reduce_f32(acc);

    __shared__ float lds_sums[NWAVES];
    int lane = threadIdx.x & 31;
    int wid  = threadIdx.x >> 5;
    if (lane == 0) lds_sums[wid] = wsum;
    __syncthreads();
    if (threadIdx.x == 0) {
        float s = 0.0f;
        #pragma unroll
        for (int i = 0; i < NWAVES; ++i) s += lds_sums[i];
        out[0] = s * inv_b;
    }
}

extern "C" void kernel_launch(void* const* d_in, const int* in_sizes, int n_in,
                              void* d_out, int out_size, void* d_ws, size_t ws_size,
                              hipStream_t stream) {
    const float* logits  = (const float*)d_in[0];   // [B, 2] f32
    const int*   targets = (const int*)d_in[1];     // [B] i32
    const float* fw      = (const float*)d_in[2];   // [B] f32
    float* out = (float*)d_out;

    const long long B = (long long)in_sizes[1];     // 16777216
    const int n_groups = (int)(B / 4);
    float* block_partials = (float*)d_ws;           // NBLOCKS floats (16 KB)

    loss_partial_kernel<<<NBLOCKS, NTHREADS, 0, stream>>>(
        logits, targets, fw, block_partials, n_groups);
    loss_finalize_kernel<<<1, NTHREADS, 0, stream>>>(
        block_partials, NBLOCKS, out, 1.0f / (float)B);
}